// LocalAttention_88622355186049
// MI455X (gfx1250) — compile-verified
//
#include <hip/hip_runtime.h>
#include <math.h>

// Problem constants (match reference)
#define B_    2
#define T_    2048
#define D_    512
#define WND   32          // WINDOW
#define WIN   63          // 2*WINDOW - 1
#define NTOK  (B_ * T_)   // 4096
#define QKV_N (3 * D_)    // 1536
#define KDIM  512         // reduction dim for BOTH GEMMs

typedef float v2f __attribute__((ext_vector_type(2)));
typedef float v4f __attribute__((ext_vector_type(4)));
typedef float v8f __attribute__((ext_vector_type(8)));

#define WMMA_F32(a, b, c) \
    __builtin_amdgcn_wmma_f32_16x16x4_f32(false, (a), false, (b), (short)0, (c), false, false)

// C(M,N) = A(M,K) @ W(N,K)^T + bias(N),  K = KDIM = 512.
// One wave computes a 32(M) x 64(N) tile: 8 accumulators,
// 2 A-fragments + 4 B-fragments per k-step (6 b64 loads -> 8 WMMAs),
// software-pipelined one k-step ahead so WMMAs overlap the loads.
//
// Fragment layouts (ISA 7.12.2, wave32):
//  A 16x4 f32 : lane<16 -> K={0,1} in regs {0,1}; lane>=16 -> K={2,3}; row = lane&15
//  B 4x16 f32 : same K striping; col = lane&15
//  C/D 16x16  : reg r, half h -> row r + 8h; col = lane&15
__global__ void __launch_bounds__(256)
gemm_bias_wmma(const float* __restrict__ A,
               const float* __restrict__ W,
               const float* __restrict__ bias,
               float* __restrict__ C,
               int M, int N)
{
    const int wave    = blockIdx.x * (blockDim.x >> 5) + (threadIdx.x >> 5);
    const int lane    = threadIdx.x & 31;
    const int nGroups = N >> 6;                // 64-wide N groups
    const int m0      = (wave / nGroups) << 5; // 32-row M tiles
    const int n0      = (wave % nGroups) << 6;
    if (m0 >= M) return;                       // wave-uniform; grids divide exactly

    const int l15 = lane & 15;
    const int h   = lane >> 4;                 // lane half: 0 or 1
    const int kh  = h << 1;                    // K sub-offset: 0 or 2

    const float* aP0 = A + (size_t)(m0 +  0 + l15) * KDIM + kh;
    const float* aP1 = A + (size_t)(m0 + 16 + l15) * KDIM + kh;
    const float* wP0 = W + (size_t)(n0 +  0 + l15) * KDIM + kh;
    const float* wP1 = W + (size_t)(n0 + 16 + l15) * KDIM + kh;
    const float* wP2 = W + (size_t)(n0 + 32 + l15) * KDIM + kh;
    const float* wP3 = W + (size_t)(n0 + 48 + l15) * KDIM + kh;

    v8f acc[8] = {};

    // Pipeline prologue: fragments for k = 0
    v2f a0 = *(const v2f*)(aP0);
    v2f a1 = *(const v2f*)(aP1);
    v2f b0 = *(const v2f*)(wP0);
    v2f b1 = *(const v2f*)(wP1);
    v2f b2 = *(const v2f*)(wP2);
    v2f b3 = *(const v2f*)(wP3);

#pragma unroll 4
    for (int k = 0; k < KDIM - 4; k += 4) {
        // Prefetch next k-step (independent of the WMMAs below)
        v2f a0n = *(const v2f*)(aP0 + k + 4);
        v2f a1n = *(const v2f*)(aP1 + k + 4);
        v2f b0n = *(const v2f*)(wP0 + k + 4);
        v2f b1n = *(const v2f*)(wP1 + k + 4);
        v2f b2n = *(const v2f*)(wP2 + k + 4);
        v2f b3n = *(const v2f*)(wP3 + k + 4);

        acc[0] = WMMA_F32(a0, b0, acc[0]);
        acc[1] = WMMA_F32(a0, b1, acc[1]);
        acc[2] = WMMA_F32(a0, b2, acc[2]);
        acc[3] = WMMA_F32(a0, b3, acc[3]);
        acc[4] = WMMA_F32(a1, b0, acc[4]);
        acc[5] = WMMA_F32(a1, b1, acc[5]);
        acc[6] = WMMA_F32(a1, b2, acc[6]);
        acc[7] = WMMA_F32(a1, b3, acc[7]);

        a0 = a0n; a1 = a1n;
        b0 = b0n; b1 = b1n; b2 = b2n; b3 = b3n;
    }
    // Pipeline epilogue: last k-step
    acc[0] = WMMA_F32(a0, b0, acc[0]);
    acc[1] = WMMA_F32(a0, b1, acc[1]);
    acc[2] = WMMA_F32(a0, b2, acc[2]);
    acc[3] = WMMA_F32(a0, b3, acc[3]);
    acc[4] = WMMA_F32(a1, b0, acc[4]);
    acc[5] = WMMA_F32(a1, b1, acc[5]);
    acc[6] = WMMA_F32(a1, b2, acc[6]);
    acc[7] = WMMA_F32(a1, b3, acc[7]);

    // Store D + bias. For acc[mt*4 + t]: row = m0 + 16*mt + r + 8h, col = n0 + 16t + l15.
#pragma unroll
    for (int mt = 0; mt < 2; ++mt) {
#pragma unroll
        for (int t = 0; t < 4; ++t) {
            const int col  = n0 + (t << 4) + l15;
            const float bn = bias[col];
            const v8f av   = acc[mt * 4 + t];
#pragma unroll
            for (int r = 0; r < 8; ++r) {
                C[(size_t)(m0 + (mt << 4) + r + 8 * h) * N + col] = av[r] + bn;
            }
        }
    }
}

// One wave32 per token: 63-wide local attention, numerically stable softmax.
// qkv rows are 1536 floats: q [0,512), k [512,1024), v [1024,1536).
__global__ void __launch_bounds__(256)
local_attn(const float* __restrict__ qkv, float* __restrict__ out)
{
    const int token = blockIdx.x * (blockDim.x >> 5) + (threadIdx.x >> 5);
    const int lane  = threadIdx.x & 31;
    if (token >= NTOK) return;

    const int t       = token & (T_ - 1);
    const int rowBase = token - t;           // b * T
    const float inv_scale = 0.044194173824159216f;  // 1/sqrt(512)

    // Lane owns dims [16*lane, 16*lane+16)
    const float* qrow = qkv + (size_t)token * QKV_N + (lane << 4);
    v4f q0 = *(const v4f*)(qrow + 0);
    v4f q1 = *(const v4f*)(qrow + 4);
    v4f q2 = *(const v4f*)(qrow + 8);
    v4f q3 = *(const v4f*)(qrow + 12);

    // Pass 1: scores; parked two-per-lane (s0: j = lane, s1: j = 32 + lane)
    float s0 = -INFINITY, s1 = -INFINITY;
    for (int j = 0; j < WIN; ++j) {
        const int jj    = t + j - (WND - 1);
        const bool ok   = (jj >= 0) && (jj < T_);
        const int  jc   = min(max(jj, 0), T_ - 1);
        const float* kr = qkv + (size_t)(rowBase + jc) * QKV_N + D_ + (lane << 4);
        v4f k0 = *(const v4f*)(kr + 0);
        v4f k1 = *(const v4f*)(kr + 4);
        v4f k2 = *(const v4f*)(kr + 8);
        v4f k3 = *(const v4f*)(kr + 12);
        float d = q0.x*k0.x + q0.y*k0.y + q0.z*k0.z + q0.w*k0.w
                + q1.x*k1.x + q1.y*k1.y + q1.z*k1.z + q1.w*k1.w
                + q2.x*k2.x + q2.y*k2.y + q2.z*k2.z + q2.w*k2.w
                + q3.x*k3.x + q3.y*k3.y + q3.z*k3.z + q3.w*k3.w;
#pragma unroll
        for (int off = 16; off; off >>= 1) d += __shfl_xor(d, off, 32);
        const float sc = ok ? d * inv_scale : -INFINITY;
        if (j < 32) { if (lane == j)      s0 = sc; }
        else        { if (lane == j - 32) s1 = sc; }
    }

    // Softmax over the 63 distributed scores
    float mx = fmaxf(s0, s1);
#pragma unroll
    for (int off = 16; off; off >>= 1) mx = fmaxf(mx, __shfl_xor(mx, off, 32));
    float e0 = expf(s0 - mx);   // -inf -> 0
    float e1 = expf(s1 - mx);
    float sum = e0 + e1;
#pragma unroll
    for (int off = 16; off; off >>= 1) sum += __shfl_xor(sum, off, 32);
    const float inv = 1.0f / sum;
    const float p0 = e0 * inv;
    const float p1 = e1 * inv;

    // Pass 2: out = sum_j p_j * v_j  (invalid j already have p == 0)
    v4f o0 = {}, o1 = {}, o2 = {}, o3 = {};
    for (int j = 0; j < WIN; ++j) {
        const int jj  = t + j - (WND - 1);
        const int jc  = min(max(jj, 0), T_ - 1);
        const float ps = (j < 32) ? p0 : p1;
        const float pj = __shfl(ps, j & 31, 32);
        const float* vr = qkv + (size_t)(rowBase + jc) * QKV_N + 2 * D_ + (lane << 4);
        v4f w0 = *(const v4f*)(vr + 0);
        v4f w1 = *(const v4f*)(vr + 4);
        v4f w2 = *(const v4f*)(vr + 8);
        v4f w3 = *(const v4f*)(vr + 12);
        o0 += pj * w0; o1 += pj * w1; o2 += pj * w2; o3 += pj * w3;
    }

    float* orow = out + (size_t)token * D_ + (lane << 4);
    *(v4f*)(orow + 0)  = o0;
    *(v4f*)(orow + 4)  = o1;
    *(v4f*)(orow + 8)  = o2;
    *(v4f*)(orow + 12) = o3;
}

extern "C" void kernel_launch(void* const* d_in, const int* in_sizes, int n_in,
                              void* d_out, int out_size, void* d_ws, size_t ws_size,
                              hipStream_t stream)
{
    (void)in_sizes; (void)n_in; (void)out_size; (void)ws_size;
    const float* x      = (const float*)d_in[0];   // (B,T,512)
    const float* w_qkv  = (const float*)d_in[1];   // (1536,512)
    const float* b_qkv  = (const float*)d_in[2];   // (1536)
    const float* w_proj = (const float*)d_in[3];   // (512,512)
    const float* b_proj = (const float*)d_in[4];   // (512)
    float* out = (float*)d_out;                    // (B,T,512)

    float* qkv  = (float*)d_ws;                          // 4096*1536 f32 = 24 MiB
    float* attn = qkv + (size_t)NTOK * QKV_N;            // 4096*512  f32 =  8 MiB

    // QKV GEMM: waves = (4096/32)*(1536/64) = 3072 -> 384 blocks of 8 waves
    gemm_bias_wmma<<<384, 256, 0, stream>>>(x, w_qkv, b_qkv, qkv, NTOK, QKV_N);

    // Attention: 4096 tokens, 8 waves/block -> 512 blocks
    local_attn<<<NTOK / 8, 256, 0, stream>>>(qkv, attn);

    // Proj GEMM: waves = (4096/32)*(512/64) = 1024 -> 128 blocks
    gemm_bias_wmma<<<128, 256, 0, stream>>>(attn, w_proj, b_proj, out, NTOK, D_);
}